// loss_func_21517786152966
// MI455X (gfx1250) — compile-verified
//
#include <hip/hip_runtime.h>

// MI455X / gfx1250, wave32. Memory-bound fused CE+MSE loss:
//   ~96 MB streamed once at 23.3 TB/s => ~4us floor; compute is negligible.
// Strategy: 4 rows (= 20 floats = 5x float4, 16B-aligned) per thread,
// fast v_exp_f32/v_log_f32, deterministic tree reduction whose 16-partial
// block-collapse is a single exact V_WMMA_F32_16X16X4_F32 (B = ones).

typedef __attribute__((ext_vector_type(2))) float v2f;
typedef __attribute__((ext_vector_type(8))) float v8f;

#define ALPHA_W 1.0f
#define BETA_W  0.5f
#define NCLS    5

__device__ __forceinline__ void row_terms(float x0, float x1, float x2,
                                          float x3, float x4, int lab,
                                          float& ce, float& mse) {
  // log-sum-exp over 5 classes (numerically stable)
  float m = fmaxf(fmaxf(fmaxf(x0, x1), fmaxf(x2, x3)), x4);
  float s = __expf(x0 - m) + __expf(x1 - m) + __expf(x2 - m) +
            __expf(x3 - m) + __expf(x4 - m);
  float lse = m + __logf(s);
  float xs = x0;
  xs = (lab == 1) ? x1 : xs;
  xs = (lab == 2) ? x2 : xs;
  xs = (lab == 3) ? x3 : xs;
  xs = (lab == 4) ? x4 : xs;
  ce = lse - xs;
  // temp = sum_j j * x[j]
  float temp = fmaf(4.0f, x4, fmaf(3.0f, x3, fmaf(2.0f, x2, x1)));
  float d = temp - (float)lab;
  mse = d * d;
}

// Wave32 shfl reduce -> 8 ce + 8 mse wave sums in LDS -> one
// V_WMMA_F32_16X16X4_F32 collapses all 16 partials (exact: B == 1.0).
// Returns true on the single lane holding (ceTot, mseTot).
__device__ __forceinline__ bool block_reduce_wmma(float ce, float mse,
                                                  float& ceTot, float& mseTot) {
#pragma unroll
  for (int off = 16; off > 0; off >>= 1) {
    ce  += __shfl_xor(ce,  off, 32);
    mse += __shfl_xor(mse, off, 32);
  }
  __shared__ float vals[16];
  const int lane = threadIdx.x & 31;
  const int wave = threadIdx.x >> 5;  // 8 waves per 256-thread block
  if (lane == 0) { vals[wave] = ce; vals[8 + wave] = mse; }
  __syncthreads();

  bool winner = false;
  if (threadIdx.x < 32) {  // wave 0 only; EXEC all-1s inside the wave
    // A (16x4 f32) layout: lanes 0-15 -> {K0,K1}, lanes 16-31 -> {K2,K3}.
    // Row 0: ce[0..3], row 1: ce[4..7], row 2: mse[0..3], row 3: mse[4..7].
    const int m  = lane & 15;
    const int hi = lane >> 4;
    v2f a;
    a.x = (m < 4) ? vals[m * 4 + 2 * hi]     : 0.0f;
    a.y = (m < 4) ? vals[m * 4 + 2 * hi + 1] : 0.0f;
    v2f b;
    b.x = 1.0f;
    b.y = 1.0f;  // all-ones B: D[m][n] = sum_k A[m][k]
    v8f c = {};
    v8f d = __builtin_amdgcn_wmma_f32_16x16x4_f32(
        /*neg_a=*/false, a, /*neg_b=*/false, b,
        /*c_mod=*/(short)0, c, /*reuse_a=*/false, /*reuse_b=*/false);
    // D VGPR j, lanes 0-15: row M=j replicated over columns.
    ceTot  = d[0] + d[1];
    mseTot = d[2] + d[3];
    winner = (lane == 0);
  }
  return winner;
}

__global__ __launch_bounds__(256) void ce_mse_partial_kernel(
    const float* __restrict__ x, const int* __restrict__ y,
    float* __restrict__ ceOut, float* __restrict__ mseOut, int nRows) {
  const int tid = blockIdx.x * 256 + threadIdx.x;
  const long long rowBase = (long long)tid * 4;

  float ce = 0.0f, mse = 0.0f;
  if (rowBase + 3 < (long long)nRows) {
    // 4 rows = 80 bytes = five coalesced b128 loads, 16B-aligned base.
    const float4* p = (const float4*)(x + rowBase * NCLS);
    float4 f0 = p[0], f1 = p[1], f2 = p[2], f3 = p[3], f4 = p[4];
    int4 lb = ((const int4*)y)[tid];
    float c0, m0;
    row_terms(f0.x, f0.y, f0.z, f0.w, f1.x, lb.x, c0, m0); ce += c0; mse += m0;
    row_terms(f1.y, f1.z, f1.w, f2.x, f2.y, lb.y, c0, m0); ce += c0; mse += m0;
    row_terms(f2.z, f2.w, f3.x, f3.y, f3.z, lb.z, c0, m0); ce += c0; mse += m0;
    row_terms(f3.w, f4.x, f4.y, f4.z, f4.w, lb.w, c0, m0); ce += c0; mse += m0;
  } else {
    for (long long r = rowBase; r < (long long)nRows; ++r) {
      const float* xr = x + r * NCLS;
      float c0, m0;
      row_terms(xr[0], xr[1], xr[2], xr[3], xr[4], y[r], c0, m0);
      ce += c0; mse += m0;
    }
  }

  float ceTot, mseTot;
  if (block_reduce_wmma(ce, mse, ceTot, mseTot)) {
    ceOut[blockIdx.x]  = ceTot;
    mseOut[blockIdx.x] = mseTot;
  }
}

__global__ __launch_bounds__(256) void ce_mse_final_kernel(
    const float* __restrict__ ceP, const float* __restrict__ mseP, int n,
    float* __restrict__ out, float invN) {
  float ce = 0.0f, mse = 0.0f;
  for (int i = threadIdx.x; i < n; i += 256) {  // fixed order => deterministic
    ce  += ceP[i];
    mse += mseP[i];
  }
  float ceTot, mseTot;
  if (block_reduce_wmma(ce, mse, ceTot, mseTot)) {
    out[0] = (ALPHA_W * ceTot + BETA_W * mseTot) * invN;
  }
}

extern "C" void kernel_launch(void* const* d_in, const int* in_sizes, int n_in,
                              void* d_out, int out_size, void* d_ws,
                              size_t ws_size, hipStream_t stream) {
  const float* x = (const float*)d_in[0];
  const int*   y = (const int*)d_in[1];
  float* out = (float*)d_out;

  const int nRows = in_sizes[0] / NCLS;           // 4,194,304
  const int nThreads = (nRows + 3) / 4;           // 4 rows per thread
  const int block = 256;
  const int grid = (nThreads + block - 1) / block;  // 4096 blocks

  float* ws = (float*)d_ws;  // [grid] ce partials | [grid] mse partials (32 KB)
  ce_mse_partial_kernel<<<grid, block, 0, stream>>>(x, y, ws, ws + grid, nRows);
  ce_mse_final_kernel<<<1, block, 0, stream>>>(ws, ws + grid, grid, out,
                                               1.0f / (float)nRows);
}